// GAT_64055142252650
// MI455X (gfx1250) — compile-verified
//
#include <hip/hip_runtime.h>

#define TIN   128   // IN_CH
#define TOUT  128   // HEADS * OUT_CH
#define NHEADS  4
#define NEG_SLOPE 0.2f

typedef __attribute__((ext_vector_type(2))) float v2f;
typedef __attribute__((ext_vector_type(8))) float v8f;

// ---------------- init: out = bias, amax = -inf, denom = 0 ----------------
__global__ __launch_bounds__(256) void gat_init_kernel(float* __restrict__ out,
                                                       const float* __restrict__ bias,
                                                       float* __restrict__ amax,
                                                       float* __restrict__ denom, int N) {
  long i = (long)blockIdx.x * blockDim.x + threadIdx.x;
  if (i < (long)N * TOUT) out[i] = bias[i & (TOUT - 1)];
  if (i < (long)N * NHEADS) {
    amax[i] = __int_as_float(0xff800000);  // -inf
    denom[i] = 0.f;
  }
}

// ---------------- GEMM: h = x @ W via V_WMMA_F32_16X16X4_F32 ----------------
// Each wave computes a 16-row strip of h across all 128 output columns
// (8 tiles of 16x16), accumulating K=128 in 32 steps of K=4.
__global__ __launch_bounds__(256) void gat_gemm_kernel(const float* __restrict__ x,
                                                       const float* __restrict__ W,
                                                       float* __restrict__ h, int N) {
  const int wave = threadIdx.x >> 5;   // 0..7
  const int lane = threadIdx.x & 31;
  const int lo = lane & 15;            // M index (A) / N index (B, C/D)
  const int hi = lane >> 4;            // K-half selector
  const int row0 = (blockIdx.x * 8 + wave) * 16;
  if (row0 >= N) return;               // wave-uniform exit; EXEC stays all-ones

  // Preload A fragments for the whole K=128 (32 frags x float2 = 64 VGPRs).
  // ISA 16x4 f32 A layout: lanes 0-15 hold (K=0,K=1), lanes 16-31 hold (K=2,K=3).
  int arow = row0 + lo;
  if (arow >= N) arow = N - 1;                 // clamp loads; stores guarded below
  const float* xrow = x + (long)arow * TIN;
  v2f afrag[32];
#pragma unroll
  for (int kk = 0; kk < 32; ++kk) {
    afrag[kk] = *(const v2f*)(xrow + kk * 4 + 2 * hi);
  }

  for (int t = 0; t < 8; ++t) {
    const int col0 = t * 16;
    v8f acc = {};
#pragma unroll
    for (int kk = 0; kk < 32; ++kk) {
      const int kb = kk * 4 + 2 * hi;
      v2f b;
      b.x = W[(long)(kb + 0) * TOUT + col0 + lo];
      b.y = W[(long)(kb + 1) * TOUT + col0 + lo];
      acc = __builtin_amdgcn_wmma_f32_16x16x4_f32(
          /*neg_a=*/false, afrag[kk], /*neg_b=*/false, b,
          /*c_mod=*/(short)0, acc, /*reuse_a=*/false, /*reuse_b=*/false);
    }
    // C/D layout: VGPR r -> M = r + 8*hi, column = lo
#pragma unroll
    for (int r = 0; r < 8; ++r) {
      const int m = row0 + r + 8 * hi;
      if (m < N) h[(long)m * TOUT + col0 + lo] = acc[r];
    }
  }
}

// ---------------- per-(node,head) attention dots ----------------
__global__ __launch_bounds__(256) void gat_att_kernel(const float* __restrict__ h,
                                                      const float* __restrict__ att_src,
                                                      const float* __restrict__ att_dst,
                                                      float* __restrict__ a_src,
                                                      float* __restrict__ a_dst, int N) {
  int i = blockIdx.x * blockDim.x + threadIdx.x;   // over N*HEADS
  if (i >= N * NHEADS) return;
  const int n = i >> 2, hd = i & 3;
  const float4* hp = (const float4*)(h + (long)n * TOUT + hd * 32);
  const float4* as = (const float4*)(att_src + hd * 32);
  const float4* ad = (const float4*)(att_dst + hd * 32);
  float s = 0.f, d = 0.f;
#pragma unroll
  for (int j = 0; j < 8; ++j) {
    float4 hv = hp[j], a = as[j], b = ad[j];
    s += hv.x * a.x + hv.y * a.y + hv.z * a.z + hv.w * a.w;
    d += hv.x * b.x + hv.y * b.y + hv.z * b.z + hv.w * b.w;
  }
  a_src[i] = s;
  a_dst[i] = d;
}

// float atomic max via signed/unsigned integer atomics (valid incl. -inf init)
__device__ inline void atomicMaxF(float* addr, float v) {
  if (v >= 0.f)
    atomicMax((int*)addr, __float_as_int(v));
  else
    atomicMin((unsigned int*)addr, __float_as_uint(v));
}

__device__ inline void edge_sd(const long long* __restrict__ ei, int e, int E,
                               long& s, long& d) {
  if (e < E) { s = (long)ei[e]; d = (long)ei[(long)E + e]; }
  else       { s = d = e - E; }   // appended self loops
}

// ---------------- pass 1: segment max of leaky-relu logits ----------------
__global__ __launch_bounds__(256) void gat_max_kernel(const long long* __restrict__ ei,
                                                      const float* __restrict__ a_src,
                                                      const float* __restrict__ a_dst,
                                                      float* __restrict__ amax, int E, int N) {
  int e = blockIdx.x * blockDim.x + threadIdx.x;
  if (e >= E + N) return;
  long s, d;
  edge_sd(ei, e, E, s, d);
#pragma unroll
  for (int hd = 0; hd < NHEADS; ++hd) {
    float l = a_src[s * NHEADS + hd] + a_dst[d * NHEADS + hd];
    l = (l >= 0.f) ? l : NEG_SLOPE * l;
    atomicMaxF(&amax[d * NHEADS + hd], l);
  }
}

// ---------------- pass 2: segment sum of exp(logit - max) ----------------
__global__ __launch_bounds__(256) void gat_denom_kernel(const long long* __restrict__ ei,
                                                        const float* __restrict__ a_src,
                                                        const float* __restrict__ a_dst,
                                                        const float* __restrict__ amax,
                                                        float* __restrict__ denom, int E, int N) {
  int e = blockIdx.x * blockDim.x + threadIdx.x;
  if (e >= E + N) return;
  long s, d;
  edge_sd(ei, e, E, s, d);
#pragma unroll
  for (int hd = 0; hd < NHEADS; ++hd) {
    float l = a_src[s * NHEADS + hd] + a_dst[d * NHEADS + hd];
    l = (l >= 0.f) ? l : NEG_SLOPE * l;
    atomicAdd(&denom[d * NHEADS + hd], __expf(l - amax[d * NHEADS + hd]));
  }
}

// ---------------- pass 3: one wave per edge: normalize + scatter ----------------
__global__ __launch_bounds__(256) void gat_scatter_kernel(const long long* __restrict__ ei,
                                                          const float* __restrict__ h,
                                                          const float* __restrict__ a_src,
                                                          const float* __restrict__ a_dst,
                                                          const float* __restrict__ amax,
                                                          const float* __restrict__ denom,
                                                          float* __restrict__ out,
                                                          float* __restrict__ alpha_out,
                                                          int E, int N) {
  const int lane = threadIdx.x & 31;
  const int e = blockIdx.x * 8 + (threadIdx.x >> 5);   // one wave32 per edge
  if (e >= E + N) return;
  long s, d;
  edge_sd(ei, e, E, s, d);

  const int hd = lane >> 3;  // lane*4 channels -> head = (lane*4)/32
  float l = a_src[s * NHEADS + hd] + a_dst[d * NHEADS + hd];
  l = (l >= 0.f) ? l : NEG_SLOPE * l;
  const float a = __expf(l - amax[d * NHEADS + hd]) /
                  (denom[d * NHEADS + hd] + 1e-16f);

  if ((lane & 7) == 0) alpha_out[(long)e * NHEADS + hd] = a;  // lanes 0,8,16,24

  const float4 hv = *(const float4*)(h + (long)s * TOUT + lane * 4);
  float* op = out + (long)d * TOUT + lane * 4;
  atomicAdd(op + 0, hv.x * a);
  atomicAdd(op + 1, hv.y * a);
  atomicAdd(op + 2, hv.z * a);
  atomicAdd(op + 3, hv.w * a);
}

extern "C" void kernel_launch(void* const* d_in, const int* in_sizes, int n_in,
                              void* d_out, int out_size, void* d_ws, size_t ws_size,
                              hipStream_t stream) {
  const float*     x       = (const float*)d_in[0];
  const long long* ei      = (const long long*)d_in[1];   // int64 edge_index [2,E]
  const float*     W       = (const float*)d_in[2];
  const float*     att_src = (const float*)d_in[3];
  const float*     att_dst = (const float*)d_in[4];
  const float*     bias    = (const float*)d_in[5];

  const int N = in_sizes[0] / TIN;
  const int E = in_sizes[1] / 2;
  const int Etot = E + N;

  float* out       = (float*)d_out;              // [N, 128]
  float* alpha_out = out + (long)N * TOUT;       // [E+N, 4]

  float* ws    = (float*)d_ws;
  float* h     = ws;  ws += (long)N * TOUT;
  float* a_src = ws;  ws += (long)N * NHEADS;
  float* a_dst = ws;  ws += (long)N * NHEADS;
  float* amax  = ws;  ws += (long)N * NHEADS;
  float* denom = ws;

  {  // init outputs/stats
    long tot = (long)N * TOUT;
    gat_init_kernel<<<(int)((tot + 255) / 256), 256, 0, stream>>>(out, bias, amax, denom, N);
  }
  {  // dense projection (WMMA f32): 8 waves/block, 128 rows/block
    gat_gemm_kernel<<<(N + 127) / 128, 256, 0, stream>>>(x, W, h, N);
  }
  {  // per-node attention contributions
    gat_att_kernel<<<(N * NHEADS + 255) / 256, 256, 0, stream>>>(h, att_src, att_dst,
                                                                 a_src, a_dst, N);
  }
  {  // softmax statistics
    int nb = (Etot + 255) / 256;
    gat_max_kernel<<<nb, 256, 0, stream>>>(ei, a_src, a_dst, amax, E, N);
    gat_denom_kernel<<<nb, 256, 0, stream>>>(ei, a_src, a_dst, amax, denom, E, N);
  }
  {  // weighted scatter-aggregate, one wave per edge
    gat_scatter_kernel<<<(Etot + 7) / 8, 256, 0, stream>>>(ei, h, a_src, a_dst, amax,
                                                           denom, out, alpha_out, E, N);
  }
}